// NeuralizedKMeans_6030134083992
// MI455X (gfx1250) — compile-verified
//
#include <hip/hip_runtime.h>
#include <hip/hip_bf16.h>

typedef __attribute__((ext_vector_type(2))) float v2f;
typedef __attribute__((ext_vector_type(8))) float v8f;

#define D_DIM 1024
#define K_DIM 32

// Kernel 1: csq[k] = ||centroid_k||^2   (K=32 blocks, 256 threads each)
__global__ void __launch_bounds__(256)
csq_kernel(const float* __restrict__ C, float* __restrict__ csq) {
    const int k = blockIdx.x;
    float s = 0.f;
    for (int d = threadIdx.x; d < D_DIM; d += 256) {
        float v = C[(size_t)k * D_DIM + d];
        s = fmaf(v, v, s);
    }
    // wave32 reduction
    #pragma unroll
    for (int mask = 16; mask >= 1; mask >>= 1)
        s += __shfl_xor(s, mask, 32);
    __shared__ float red[8];
    if ((threadIdx.x & 31) == 0) red[threadIdx.x >> 5] = s;
    __syncthreads();
    if (threadIdx.x == 0) {
        float t = 0.f;
        #pragma unroll
        for (int i = 0; i < 8; ++i) t += red[i];
        csq[k] = t;
    }
}

// Kernel 2: per row n compute G[n,k] = x_n . mu_k via V_WMMA_F32_16X16X4_F32,
// d_k = csq[k] - 2*G[n,k], out[n] = secondmin(d) - min(d).
__global__ void __launch_bounds__(256)
margin_kernel(const float* __restrict__ X, const float* __restrict__ C,
              const float* __restrict__ csq, float* __restrict__ out, int N) {
    const int lane = threadIdx.x & 31;
    const int wave = threadIdx.x >> 5;       // 8 waves per block
    const int m    = lane & 15;              // column within 16-wide tile / A row
    const int half = lane >> 4;              // 0: K=0..1, 1: K=2..3 of the k-chunk

    const int row_base = blockIdx.x * 128 + wave * 16;

    // A fragment source: 16 rows of X, clamped for the ragged tail
    int arow = row_base + m;
    if (arow >= N) arow = N - 1;
    const float* aptr  = X + (size_t)arow * D_DIM + 2 * half;
    // B fragment sources: column tiles 0 (cols 0..15) and 1 (cols 16..31)
    const float* bptr0 = C + (size_t)m        * D_DIM + 2 * half;
    const float* bptr1 = C + (size_t)(16 + m) * D_DIM + 2 * half;

    v8f acc0 = {};
    v8f acc1 = {};

    for (int d0 = 0; d0 < D_DIM; d0 += 64) {
        // stream-prefetch X ~1KB ahead (speculative; OOB drops silently)
        __builtin_prefetch(aptr + d0 + 256, 0, 0);
        #pragma unroll
        for (int dd = 0; dd < 64; dd += 4) {
            const int d = d0 + dd;
            v2f a  = *(const v2f*)(aptr  + d);
            v2f b0 = *(const v2f*)(bptr0 + d);
            v2f b1 = *(const v2f*)(bptr1 + d);
            acc0 = __builtin_amdgcn_wmma_f32_16x16x4_f32(
                false, a, false, b0, (short)0, acc0, false, false);
            acc1 = __builtin_amdgcn_wmma_f32_16x16x4_f32(
                false, a, false, b1, (short)0, acc1, false, false);
        }
    }

    const float c0 = csq[m];
    const float c1 = csq[16 + m];

    // C/D layout: acc[vg] holds row (vg + 8*half), col = m (+16 for tile 1).
    #pragma unroll
    for (int vg = 0; vg < 8; ++vg) {
        float v0 = fmaf(-2.f, acc0[vg], c0);
        float v1 = fmaf(-2.f, acc1[vg], c1);
        float m1 = fminf(v0, v1);   // smallest
        float m2 = fmaxf(v0, v1);   // second smallest
        // butterfly over the 16-lane half: merge (min, second-min) pairs
        #pragma unroll
        for (int mask = 1; mask <= 8; mask <<= 1) {
            float o1  = __shfl_xor(m1, mask, 32);
            float o2  = __shfl_xor(m2, mask, 32);
            float nm1 = fminf(m1, o1);
            m2 = fminf(fmaxf(m1, o1), fminf(m2, o2));
            m1 = nm1;
        }
        if (m == 0) {               // lanes 0 and 16 hold rows vg and vg+8
            int r = row_base + vg + 8 * half;
            if (r < N) out[r] = m2 - m1;   // d_(2) - d_(1)
        }
    }
}

extern "C" void kernel_launch(void* const* d_in, const int* in_sizes, int n_in,
                              void* d_out, int out_size, void* d_ws, size_t ws_size,
                              hipStream_t stream) {
    const float* X   = (const float*)d_in[0];   // [N, 1024] f32
    const float* C   = (const float*)d_in[1];   // [32, 1024] f32
    float*       out = (float*)d_out;           // [N] f32
    float*       csq = (float*)d_ws;            // 32 floats of scratch

    const int N = in_sizes[0] / D_DIM;

    csq_kernel<<<K_DIM, 256, 0, stream>>>(C, csq);

    const int blocks = (N + 127) / 128;         // 128 rows per block (8 waves x 16)
    margin_kernel<<<blocks, 256, 0, stream>>>(X, C, csq, out, N);
}